// EuclideanCodebook_72361609003145
// MI455X (gfx1250) — compile-verified
//
#include <hip/hip_runtime.h>

typedef __attribute__((ext_vector_type(2))) float    v2f;
typedef __attribute__((ext_vector_type(8))) float    v8f;
typedef __attribute__((ext_vector_type(4))) unsigned v4u;
typedef __attribute__((ext_vector_type(8))) int      v8i;
typedef __attribute__((ext_vector_type(4))) int      v4i;

#define D_DIM       256
#define CHUNK       64                      // codebook columns staged per TDM load
#define LDS_STRIDE  260                     // 256 + 4-DWORD pad -> conflict-free ds_load_b64
#define BUF_FLOATS  (CHUNK * LDS_STRIDE)    // 16640 floats = 66560 B per buffer
#define LDS_BYTES   (2 * BUF_FLOATS * 4)    // 133120 B double-buffered

// ---------------------------------------------------------------------------
// Kernel 1: c2[k] = sum_d embed[k][d]^2   (one wave32 per codebook row)
// ---------------------------------------------------------------------------
__global__ __launch_bounds__(256) void codebook_norms(const float* __restrict__ embed,
                                                      float* __restrict__ c2, int K) {
  int wave = (blockIdx.x * blockDim.x + threadIdx.x) >> 5;
  int lane = threadIdx.x & 31;
  if (wave >= K) return;
  const float* row = embed + (size_t)wave * D_DIM;
  float s = 0.f;
#pragma unroll
  for (int i = 0; i < D_DIM / 32; ++i) {
    float v = row[lane + i * 32];
    s = __builtin_fmaf(v, v, s);
  }
#pragma unroll
  for (int off = 16; off; off >>= 1) s += __shfl_xor(s, off, 32);
  if (lane == 0) c2[wave] = s;
}

// ---------------------------------------------------------------------------
// TDM: stage a 64-row x 256-float embed chunk into LDS.
// D# built per CDNA5 ISA 8.3/8.4: 2D tensor, data_size=4B, tile 256x64,
// pad_enable with interval=256 DWORDs, amount=4 DWORDs (LDS row stride 260).
// ---------------------------------------------------------------------------
__device__ __forceinline__ void tdm_load_chunk(const float* embed, int n0,
                                               unsigned lds_byte_off) {
  unsigned long long ga =
      (unsigned long long)(uintptr_t)(embed + (size_t)n0 * D_DIM);
  v4u g0;
  g0[0] = 1u;                                              // count=1 (valid), user mode
  g0[1] = lds_byte_off;                                    // lds_addr (bytes)
  g0[2] = (unsigned)(ga & 0xffffffffu);                    // global_addr[31:0]
  g0[3] = (unsigned)((ga >> 32) & 0x01ffffffu) | (2u << 30); // ga[56:32], type=2
  v8i g1;
  g1[0] = (int)((2u << 16)       // data_size = 4 bytes
              | (1u << 20)       // pad_enable
              | (7u << 22)       // pad_interval: every 256 DWORDs
              | (3u << 25));     // pad_amount: 4 DWORDs
  g1[1] = (int)(256u << 16);     // tensor_dim0 = 256
  g1[2] = (int)((unsigned)CHUNK << 16); // tensor_dim1 = 64
  g1[3] = (int)(256u << 16);     // tile_dim0 = 256
  g1[4] = CHUNK;                 // tile_dim1 = 64, tile_dim2 = 0 (2D)
  g1[5] = 256;                   // tensor_dim0_stride = 256 elements
  g1[6] = 0;
  g1[7] = 0;
  v4i gz4 = {0, 0, 0, 0};
  v8i gz8 = {0, 0, 0, 0, 0, 0, 0, 0};
  __builtin_amdgcn_tensor_load_to_lds(g0, g1, gz4, gz4, gz8, 0);
}

// ---------------------------------------------------------------------------
// Kernel 2: fused GEMM (x . embed^T via V_WMMA_F32_16X16X4_F32) + running
// argmin of (c2[k] - 2 x.c).  Embed chunks are TDM-staged to LDS once per
// workgroup (8x VMEM reduction), double-buffered on TENSORcnt; B operands
// come from bank-conflict-free LDS, A operands stream from global (L0-hot).
// ---------------------------------------------------------------------------
__global__ __launch_bounds__(256) void vq_argmin(const float* __restrict__ x,
                                                 const float* __restrict__ embed,
                                                 const float* __restrict__ c2,
                                                 int* __restrict__ out_ind,
                                                 int N, int K) {
  extern __shared__ float smem[];
  const int lane = threadIdx.x & 31;
  const int wave = threadIdx.x >> 5;
  const int m0   = (blockIdx.x * 8 + wave) * 16;   // 16 rows per wave
  const int half = lane >> 4;
  const int l15  = lane & 15;
  const int koff = half * 2;                       // K-slot pair (ISA A/B layout)

  const float* xrow = x + (size_t)(m0 + l15) * D_DIM;

  float minval[8];
  int   minidx[8];
#pragma unroll
  for (int j = 0; j < 8; ++j) { minval[j] = 3.4e38f; minidx[j] = 0; }

  const int NCHUNK = K / CHUNK;
  if (wave == 0) {
    tdm_load_chunk(embed, 0, 0u);
    tdm_load_chunk(embed, CHUNK, (unsigned)(BUF_FLOATS * 4));
  }

  for (int c = 0; c < NCHUNK; ++c) {
    const int n0 = c * CHUNK;
    if (wave == 0) {
      if (c + 1 < NCHUNK) __builtin_amdgcn_s_wait_tensorcnt(1);
      else                __builtin_amdgcn_s_wait_tensorcnt(0);
    }
    __syncthreads();   // chunk c resident in buffer (c&1) for all waves

    const float* bb = smem + (size_t)(c & 1) * BUF_FLOATS;
    const float* b0p = bb + (size_t)(l15 +  0) * LDS_STRIDE + koff;
    const float* b1p = bb + (size_t)(l15 + 16) * LDS_STRIDE + koff;
    const float* b2p = bb + (size_t)(l15 + 32) * LDS_STRIDE + koff;
    const float* b3p = bb + (size_t)(l15 + 48) * LDS_STRIDE + koff;

    v8f acc0 = {}, acc1 = {}, acc2 = {}, acc3 = {};
#pragma unroll 8
    for (int d = 0; d < D_DIM; d += 4) {
      v2f a  = *(const v2f*)(xrow + d + koff);
      v2f b0 = *(const v2f*)(b0p + d);
      v2f b1 = *(const v2f*)(b1p + d);
      v2f b2 = *(const v2f*)(b2p + d);
      v2f b3 = *(const v2f*)(b3p + d);
      acc0 = __builtin_amdgcn_wmma_f32_16x16x4_f32(false, a, false, b0, (short)0, acc0, false, false);
      acc1 = __builtin_amdgcn_wmma_f32_16x16x4_f32(false, a, false, b1, (short)0, acc1, false, false);
      acc2 = __builtin_amdgcn_wmma_f32_16x16x4_f32(false, a, false, b2, (short)0, acc2, false, false);
      acc3 = __builtin_amdgcn_wmma_f32_16x16x4_f32(false, a, false, b3, (short)0, acc3, false, false);
    }

    // score = c2[n] - 2 * (x . c); ascending-k + strict < matches argmin ties
    float c20 = c2[n0 +  0 + l15];
    float c21 = c2[n0 + 16 + l15];
    float c22 = c2[n0 + 32 + l15];
    float c23 = c2[n0 + 48 + l15];
#pragma unroll
    for (int j = 0; j < 8; ++j) {
      float v0 = __builtin_fmaf(-2.f, acc0[j], c20);
      float v1 = __builtin_fmaf(-2.f, acc1[j], c21);
      float v2 = __builtin_fmaf(-2.f, acc2[j], c22);
      float v3 = __builtin_fmaf(-2.f, acc3[j], c23);
      if (v0 < minval[j]) { minval[j] = v0; minidx[j] = n0 +  0 + l15; }
      if (v1 < minval[j]) { minval[j] = v1; minidx[j] = n0 + 16 + l15; }
      if (v2 < minval[j]) { minval[j] = v2; minidx[j] = n0 + 32 + l15; }
      if (v3 < minval[j]) { minval[j] = v3; minidx[j] = n0 + 48 + l15; }
    }

    __syncthreads();   // all waves done reading buffer (c&1)
    if (wave == 0 && c + 2 < NCHUNK) {
      tdm_load_chunk(embed, (c + 2) * CHUNK, (unsigned)((c & 1) * (BUF_FLOATS * 4)));
    }
  }

  // Cross-lane argmin within each 16-lane half (row m = m0 + 8*half + j).
#pragma unroll
  for (int j = 0; j < 8; ++j) {
    float v  = minval[j];
    int   id = minidx[j];
#pragma unroll
    for (int off = 1; off < 16; off <<= 1) {
      float ov  = __shfl_xor(v, off, 32);
      int   oid = __shfl_xor(id, off, 32);
      if (ov < v || (ov == v && oid < id)) { v = ov; id = oid; }
    }
    if (l15 == 0) out_ind[m0 + 8 * half + j] = id;
  }
}

// ---------------------------------------------------------------------------
// Kernel 3: quantize[n] = embed[ind[n]]   (one block per row, float4 copies)
// ---------------------------------------------------------------------------
__global__ __launch_bounds__(64) void vq_gather(const float* __restrict__ embed,
                                                const int* __restrict__ ind,
                                                float* __restrict__ q) {
  int row = blockIdx.x;
  int t   = threadIdx.x;  // 64 threads * float4 = 256 floats
  int id  = ind[row];
  const float4* src = (const float4*)(embed + (size_t)id * D_DIM);
  float4*       dst = (float4*)(q + (size_t)row * D_DIM);
  dst[t] = src[t];
}

// ---------------------------------------------------------------------------
extern "C" void kernel_launch(void* const* d_in, const int* in_sizes, int n_in,
                              void* d_out, int out_size, void* d_ws, size_t ws_size,
                              hipStream_t stream) {
  const float* x     = (const float*)d_in[0];
  const float* embed = (const float*)d_in[1];
  const int D = D_DIM;
  const int N = in_sizes[0] / D;  // 32768
  const int K = in_sizes[1] / D;  // 8192

  float* c2  = (float*)d_ws;                          // K floats of scratch
  float* q   = (float*)d_out;                         // [N, D] quantize
  int*   ind = (int*)((float*)d_out + (size_t)N * D); // [N] indices (int32 bits)

  codebook_norms<<<K / 8, 256, 0, stream>>>(embed, c2, K);
  vq_argmin<<<N / 128, 256, LDS_BYTES, stream>>>(x, embed, c2, ind, N, K);
  vq_gather<<<N, 64, 0, stream>>>(embed, ind, q);
}